// PPIModel2_6957847020277
// MI455X (gfx1250) — compile-verified
//
#include <hip/hip_runtime.h>
#include <hip/hip_bf16.h>
#include <math.h>

// GAT (2-layer) for MI455X / gfx1250.
// Dense GEMMs use V_WMMA_F32_16X16X4_F32 (one wave per 16-node tile).
// Edge softmax/aggregation uses per-node precomputed attention scalars
// (p = z . a_src, q = z . a_dst) so edge passes gather only 2 floats/edge/head.
// Workspace layout (floats), N = #nodes:
//   z1   [2][N][16]   @ 0       (32N)
//   p1   [2][N]       @ 32N     (2N)
//   q1   [2][N]       @ 34N     (2N)
//   m1   [2][N]       @ 36N     (2N)   (init -inf)
//   den1 [2][N]       @ 38N     (2N)   (init 0)
//   x    [N][32]      @ 40N     (32N)  (agg1 -> elu, layer-2 input)
//   z2   [N][2]       @ 72N     (2N)
//   p2   [N]          @ 74N
//   q2   [N]          @ 75N
//   m2   [N]          @ 76N     (init -inf)
//   den2 [N]          @ 77N     (init 0)
//   agg2 [N][2]       @ 78N     (2N)   (init 0)
//   red  [1]          @ 80N     (init 0)
// Total 80N+1 floats (~61 MB for N=200000).

typedef __attribute__((ext_vector_type(2))) float v2f;
typedef __attribute__((ext_vector_type(8))) float v8f;

#define NEG_INF (-3.0e38f)

__device__ __forceinline__ float leaky01(float e) { return e > 0.f ? e : 0.01f * e; }

// Atomic float max via monotone int-bit ordering (init value is negative -inf).
__device__ __forceinline__ void atomicMaxF(float* addr, float val) {
    if (val >= 0.f)
        atomicMax((int*)addr, __float_as_int(val));
    else
        atomicMin((unsigned int*)addr, (unsigned int)__float_as_int(val));
}

__global__ void fill_k(float* __restrict__ p, float v, int n) {
    int i = blockIdx.x * blockDim.x + threadIdx.x;
    if (i < n) p[i] = v;
}

// ---------------- Layer 1 GEMM: z1[h] = h @ fc1[h]  (N x 2 @ 2 x 16) ----------------
// One wave per 16-row tile. A = 16x4 f32 (K padded 2->4 with zeros), B = 4x16.
// A layout: lanes 0-15 hold {A[m][0],A[m][1]}, lanes 16-31 hold {A[m][2],A[m][3]} (zeros).
// B layout: lanes 0-15 hold {B[0][n],B[1][n]}, lanes 16-31 hold {B[2][n],B[3][n]} (zeros).
// All loads are unconditional (addresses clamped in-bounds) + cndmask select, so no
// per-lane EXEC save/restore around the WMMA.
__global__ void l1_wmma_k(const float* __restrict__ h, const float* __restrict__ fc1,
                          float* __restrict__ z1, int N, int ntiles) {
    int wave = (blockIdx.x * blockDim.x + threadIdx.x) >> 5;
    if (wave >= ntiles) return;           // wave-uniform: EXEC stays all-ones for WMMA
    int lane = threadIdx.x & 31;
    int l = lane & 15, half = lane >> 4;
    int m = wave * 16 + l;
    int ma = m < N ? m : 0;               // clamped, always in-bounds

    float2 hrow = *(const float2*)(h + (size_t)ma * 2);
    bool a_ok = (half == 0) && (m < N);
    v2f a;
    a.x = a_ok ? hrow.x : 0.f;
    a.y = a_ok ? hrow.y : 0.f;

    bool full = (wave * 16 + 16) <= N;    // wave-uniform

    #pragma unroll
    for (int head = 0; head < 2; ++head) {
        // fc1 is [head][in=2][out=16]; indices always in-bounds (<64)
        float b0 = fc1[head * 32 + l];
        float b1 = fc1[head * 32 + 16 + l];
        v2f b;
        b.x = (half == 0) ? b0 : 0.f;
        b.y = (half == 0) ? b1 : 0.f;

        v8f c = {};
        c = __builtin_amdgcn_wmma_f32_16x16x4_f32(false, a, false, b, (short)0, c, false, false);

        // C layout: lane holds column n = l, rows r + 8*half (r = 0..7)
        float* zb = z1 + (size_t)head * N * 16;
        int base = wave * 16 + 8 * half;
        if (full) {
            #pragma unroll
            for (int r = 0; r < 8; ++r)
                zb[(size_t)(base + r) * 16 + l] = c[r];
        } else {
            #pragma unroll
            for (int r = 0; r < 8; ++r)
                if (base + r < N) zb[(size_t)(base + r) * 16 + l] = c[r];
        }
    }
}

// p1[h][n] = z1[h][n] . attn1[h][0:16];  q1[h][n] = z1[h][n] . attn1[h][16:32]
__global__ void pq1_k(const float* __restrict__ z1, const float* __restrict__ attn1,
                      float* __restrict__ p, float* __restrict__ q, int N) {
    int n = blockIdx.x * blockDim.x + threadIdx.x;
    if (n >= N) return;
    #pragma unroll
    for (int head = 0; head < 2; ++head) {
        const float4* zr = (const float4*)(z1 + ((size_t)head * N + n) * 16);
        const float* A = attn1 + head * 32;
        float pp = 0.f, qq = 0.f;
        #pragma unroll
        for (int v = 0; v < 4; ++v) {
            float4 zv = zr[v];
            pp += zv.x * A[4 * v] + zv.y * A[4 * v + 1] + zv.z * A[4 * v + 2] + zv.w * A[4 * v + 3];
            qq += zv.x * A[16 + 4 * v] + zv.y * A[16 + 4 * v + 1] +
                  zv.z * A[16 + 4 * v + 2] + zv.w * A[16 + 4 * v + 3];
        }
        p[(size_t)head * N + n] = pp;
        q[(size_t)head * N + n] = qq;
    }
}

// ---------------- Edge passes (generic over #heads H) ----------------
template <int H>
__global__ void edge_max_k(const int* __restrict__ src, const int* __restrict__ dst,
                           const float* __restrict__ p, const float* __restrict__ q,
                           float* __restrict__ m, int E, int N) {
    int i = blockIdx.x * blockDim.x + threadIdx.x;
    if (i >= E) return;
    int s = src[i], d = dst[i];
    #pragma unroll
    for (int h = 0; h < H; ++h) {
        float e = leaky01(p[(size_t)h * N + s] + q[(size_t)h * N + d]);
        atomicMaxF(&m[(size_t)h * N + d], e);
    }
}

template <int H>
__global__ void edge_den_k(const int* __restrict__ src, const int* __restrict__ dst,
                           const float* __restrict__ p, const float* __restrict__ q,
                           const float* __restrict__ m, float* __restrict__ den,
                           int E, int N) {
    int i = blockIdx.x * blockDim.x + threadIdx.x;
    if (i >= E) return;
    int s = src[i], d = dst[i];
    #pragma unroll
    for (int h = 0; h < H; ++h) {
        float e = leaky01(p[(size_t)h * N + s] + q[(size_t)h * N + d]);
        float ex = __expf(e - m[(size_t)h * N + d]);
        atomicAdd(&den[(size_t)h * N + d], ex);
    }
}

// Layer-1 aggregation: D=16 per head, z rows gathered as 4 x float4.
__global__ void edge_agg16_k(const int* __restrict__ src, const int* __restrict__ dst,
                             const float* __restrict__ p, const float* __restrict__ q,
                             const float* __restrict__ m, const float* __restrict__ den,
                             const float* __restrict__ z, float* __restrict__ agg,
                             int E, int N) {
    int i = blockIdx.x * blockDim.x + threadIdx.x;
    if (i >= E) return;
    int s = src[i], d = dst[i];
    #pragma unroll
    for (int h = 0; h < 2; ++h) {
        float e = leaky01(p[(size_t)h * N + s] + q[(size_t)h * N + d]);
        float alpha = __expf(e - m[(size_t)h * N + d]) / den[(size_t)h * N + d];
        const float4* zr = (const float4*)(z + ((size_t)h * N + s) * 16);
        float* ag = agg + (size_t)d * 32 + h * 16;
        #pragma unroll
        for (int v = 0; v < 4; ++v) {
            float4 zv = zr[v];
            atomicAdd(&ag[4 * v + 0], alpha * zv.x);
            atomicAdd(&ag[4 * v + 1], alpha * zv.y);
            atomicAdd(&ag[4 * v + 2], alpha * zv.z);
            atomicAdd(&ag[4 * v + 3], alpha * zv.w);
        }
    }
}

// Layer-2 aggregation: single head, D=2, z row gathered as one float2.
__global__ void edge_agg2_k(const int* __restrict__ src, const int* __restrict__ dst,
                            const float* __restrict__ p, const float* __restrict__ q,
                            const float* __restrict__ m, const float* __restrict__ den,
                            const float* __restrict__ z, float* __restrict__ agg,
                            int E, int N) {
    int i = blockIdx.x * blockDim.x + threadIdx.x;
    if (i >= E) return;
    int s = src[i], d = dst[i];
    float e = leaky01(p[s] + q[d]);
    float alpha = __expf(e - m[d]) / den[d];
    float2 zv = *(const float2*)(z + (size_t)s * 2);
    atomicAdd(&agg[(size_t)d * 2 + 0], alpha * zv.x);
    atomicAdd(&agg[(size_t)d * 2 + 1], alpha * zv.y);
}

__global__ void elu_k(float* __restrict__ x, int n) {
    int i = blockIdx.x * blockDim.x + threadIdx.x;
    if (i >= n) return;
    float v = x[i];
    x[i] = v > 0.f ? v : (__expf(v) - 1.f);
}

// ---------------- Layer 2 GEMM: z2 = x @ fc2  (N x 32 @ 32 x 2) ----------------
// 8 chained V_WMMA_F32_16X16X4_F32 over K=32; B cols 2..15 zero-padded.
__global__ void l2_wmma_k(const float* __restrict__ x, const float* __restrict__ fc2,
                          float* __restrict__ z2, int N, int ntiles) {
    int wave = (blockIdx.x * blockDim.x + threadIdx.x) >> 5;
    if (wave >= ntiles) return;
    int lane = threadIdx.x & 31;
    int l = lane & 15, half = lane >> 4;
    int m = wave * 16 + l;
    int ma = m < N ? m : 0;               // clamped, always in-bounds
    bool m_ok = (m < N);

    v8f c = {};
    #pragma unroll
    for (int kk = 0; kk < 8; ++kk) {
        int k0 = kk * 4 + 2 * half;       // this lane-half supplies K = k0, k0+1
        float2 xv = *(const float2*)(x + (size_t)ma * 32 + k0);
        v2f a;
        a.x = m_ok ? xv.x : 0.f;
        a.y = m_ok ? xv.y : 0.f;
        // fc2 is [32][2]; clamp column so loads stay in-bounds, select to zero-pad
        int lc = l < 2 ? l : 0;
        float b0 = fc2[k0 * 2 + lc];
        float b1 = fc2[(k0 + 1) * 2 + lc];
        v2f b;
        b.x = (l < 2) ? b0 : 0.f;
        b.y = (l < 2) ? b1 : 0.f;
        c = __builtin_amdgcn_wmma_f32_16x16x4_f32(false, a, false, b, (short)0, c, false, false);
    }

    bool full = (wave * 16 + 16) <= N;    // wave-uniform
    if (l < 2) {                          // only columns 0,1 are real outputs
        int base = wave * 16 + 8 * half;
        if (full) {
            #pragma unroll
            for (int r = 0; r < 8; ++r)
                z2[(size_t)(base + r) * 2 + l] = c[r];
        } else {
            #pragma unroll
            for (int r = 0; r < 8; ++r)
                if (base + r < N) z2[(size_t)(base + r) * 2 + l] = c[r];
        }
    }
}

__global__ void pq2_k(const float* __restrict__ z2, const float* __restrict__ attn2,
                      float* __restrict__ p, float* __restrict__ q, int N) {
    int n = blockIdx.x * blockDim.x + threadIdx.x;
    if (n >= N) return;
    float2 zv = *(const float2*)(z2 + (size_t)n * 2);
    p[n] = zv.x * attn2[0] + zv.y * attn2[1];
    q[n] = zv.x * attn2[2] + zv.y * attn2[3];
}

__global__ void dot_k(const float* __restrict__ a, const float* __restrict__ w,
                      float* __restrict__ red, int n) {
    int i = blockIdx.x * blockDim.x + threadIdx.x;
    float v = (i < n) ? a[i] * w[i] : 0.f;
    #pragma unroll
    for (int off = 16; off > 0; off >>= 1) v += __shfl_down(v, off, 32);
    if ((threadIdx.x & 31) == 0) atomicAdd(red, v);
}

__global__ void fin_k(const float* __restrict__ red, const float* __restrict__ b,
                      float* __restrict__ out) {
    out[0] = 1.f / (1.f + __expf(-(red[0] + b[0])));
}

static inline int cdiv(int a, int b) { return (a + b - 1) / b; }

extern "C" void kernel_launch(void* const* d_in, const int* in_sizes, int n_in,
                              void* d_out, int out_size, void* d_ws, size_t ws_size,
                              hipStream_t stream) {
    const float* h     = (const float*)d_in[0];
    const int*   src   = (const int*)d_in[1];
    const int*   dst   = (const int*)d_in[2];
    const float* fc1   = (const float*)d_in[3];
    const float* attn1 = (const float*)d_in[4];
    const float* fc2   = (const float*)d_in[5];
    const float* attn2 = (const float*)d_in[6];
    const float* mlpw  = (const float*)d_in[7];
    const float* mlpb  = (const float*)d_in[8];

    const int N = in_sizes[0] / 2;
    const int E = in_sizes[1];
    const size_t Nz = (size_t)N;

    float* ws   = (float*)d_ws;
    float* z1   = ws;
    float* p1   = ws + 32 * Nz;
    float* q1   = ws + 34 * Nz;
    float* m1   = ws + 36 * Nz;
    float* den1 = ws + 38 * Nz;
    float* x    = ws + 40 * Nz;
    float* z2   = ws + 72 * Nz;
    float* p2   = ws + 74 * Nz;
    float* q2   = ws + 75 * Nz;
    float* m2   = ws + 76 * Nz;
    float* den2 = ws + 77 * Nz;
    float* agg2 = ws + 78 * Nz;
    float* red  = ws + 80 * Nz;

    const int B = 256;
    // init
    fill_k<<<cdiv(2 * N, B), B, 0, stream>>>(m1, NEG_INF, 2 * N);
    fill_k<<<cdiv(2 * N, B), B, 0, stream>>>(den1, 0.f, 2 * N);
    fill_k<<<cdiv(32 * N, B), B, 0, stream>>>(x, 0.f, 32 * N);
    fill_k<<<cdiv(N, B), B, 0, stream>>>(m2, NEG_INF, N);
    fill_k<<<cdiv(N, B), B, 0, stream>>>(den2, 0.f, N);
    fill_k<<<cdiv(2 * N, B), B, 0, stream>>>(agg2, 0.f, 2 * N);
    fill_k<<<1, 32, 0, stream>>>(red, 0.f, 1);

    // layer 1
    int tiles = cdiv(N, 16);
    int waves_per_blk = B / 32;
    l1_wmma_k<<<cdiv(tiles, waves_per_blk), B, 0, stream>>>(h, fc1, z1, N, tiles);
    pq1_k<<<cdiv(N, B), B, 0, stream>>>(z1, attn1, p1, q1, N);
    edge_max_k<2><<<cdiv(E, B), B, 0, stream>>>(src, dst, p1, q1, m1, E, N);
    edge_den_k<2><<<cdiv(E, B), B, 0, stream>>>(src, dst, p1, q1, m1, den1, E, N);
    edge_agg16_k<<<cdiv(E, B), B, 0, stream>>>(src, dst, p1, q1, m1, den1, z1, x, E, N);
    elu_k<<<cdiv(32 * N, B), B, 0, stream>>>(x, 32 * N);

    // layer 2
    l2_wmma_k<<<cdiv(tiles, waves_per_blk), B, 0, stream>>>(x, fc2, z2, N, tiles);
    pq2_k<<<cdiv(N, B), B, 0, stream>>>(z2, attn2, p2, q2, N);
    edge_max_k<1><<<cdiv(E, B), B, 0, stream>>>(src, dst, p2, q2, m2, E, N);
    edge_den_k<1><<<cdiv(E, B), B, 0, stream>>>(src, dst, p2, q2, m2, den2, E, N);
    edge_agg2_k<<<cdiv(E, B), B, 0, stream>>>(src, dst, p2, q2, m2, den2, z2, agg2, E, N);

    // final MLP + sigmoid
    dot_k<<<cdiv(2 * N, B), B, 0, stream>>>(agg2, mlpw, red, 2 * N);
    fin_k<<<1, 1, 0, stream>>>(red, mlpb, (float*)d_out);
}